// SOMLayer_66357244723413
// MI455X (gfx1250) — compile-verified
//
#include <hip/hip_runtime.h>

// Problem constants (from reference)
#define NS 2048   // samples
#define DD 256    // latent dim
#define PP 1024   // prototypes (32x32)

typedef __attribute__((ext_vector_type(2))) float v2f;
typedef __attribute__((ext_vector_type(8))) float v8f;

// ---------------------------------------------------------------------------
// Kernel 1: squared norms of inputs (x_sq, N) and prototypes (p_sq, P).
// One 256-thread block per row (row length == 256).
// ---------------------------------------------------------------------------
__global__ __launch_bounds__(256) void som_sq_norms(const float* __restrict__ X,
                                                    const float* __restrict__ Pr,
                                                    float* __restrict__ x_sq,
                                                    float* __restrict__ p_sq) {
  __shared__ float s[256];
  int row = blockIdx.x;
  const float* src;
  float* dst;
  if (row < NS) { src = X + (size_t)row * DD; dst = x_sq + row; }
  else          { src = Pr + (size_t)(row - NS) * DD; dst = p_sq + (row - NS); }
  float v = src[threadIdx.x];
  s[threadIdx.x] = v * v;
  __syncthreads();
  for (int st = 128; st > 0; st >>= 1) {
    if (threadIdx.x < (unsigned)st) s[threadIdx.x] += s[threadIdx.x + st];
    __syncthreads();
  }
  if (threadIdx.x == 0) *dst = s[0];
}

// ---------------------------------------------------------------------------
// Kernel 2: E[j,k] = exp(-dist[j,k]^2 / (2*sigma^2)), P*P elements.
// ---------------------------------------------------------------------------
__global__ __launch_bounds__(256) void som_exp(const float* __restrict__ dist,
                                               const float* __restrict__ sigma,
                                               float* __restrict__ E) {
  int i = blockIdx.x * 256 + threadIdx.x;
  float s = sigma[0];
  float inv = -1.0f / (2.0f * s * s);
  float d = dist[i];
  E[i] = __expf(d * d * inv);
}

// ---------------------------------------------------------------------------
// Kernel 3: per-row stats of E: inv_r[j] = 1/sum_k E[j,k],
//           b[j] = (sum_k E[j,k]*p_sq[k]) * inv_r[j].
// ---------------------------------------------------------------------------
__global__ __launch_bounds__(256) void som_row_stats(const float* __restrict__ E,
                                                     const float* __restrict__ p_sq,
                                                     float* __restrict__ inv_r,
                                                     float* __restrict__ bv) {
  __shared__ float sr[256];
  __shared__ float sc[256];
  int j = blockIdx.x;
  float r = 0.0f, c = 0.0f;
  for (int k = threadIdx.x; k < PP; k += 256) {
    float e = E[(size_t)j * PP + k];
    r += e;
    c += e * p_sq[k];
  }
  sr[threadIdx.x] = r;
  sc[threadIdx.x] = c;
  __syncthreads();
  for (int st = 128; st > 0; st >>= 1) {
    if (threadIdx.x < (unsigned)st) {
      sr[threadIdx.x] += sr[threadIdx.x + st];
      sc[threadIdx.x] += sc[threadIdx.x + st];
    }
    __syncthreads();
  }
  if (threadIdx.x == 0) {
    float ir = 1.0f / sr[0];
    inv_r[j] = ir;
    bv[j] = sc[0] * ir;
  }
}

// ---------------------------------------------------------------------------
// Kernel 4 (WMMA): Mt[d, j] = inv_r[j] * sum_k E[j,k] * Proto[k,d]
// A = E (P x P), B = Proto (P x D), D written transposed (D x P).
// One wave per 16x16 output tile; 4 waves per block; 1024 tiles total.
// K = 1024 -> 256 x v_wmma_f32_16x16x4_f32 per wave.
// ---------------------------------------------------------------------------
__global__ __launch_bounds__(128) void som_gemm1(const float* __restrict__ E,
                                                 const float* __restrict__ Pr,
                                                 const float* __restrict__ inv_r,
                                                 float* __restrict__ Mt) {
  int wave = threadIdx.x >> 5;
  int lane = threadIdx.x & 31;
  int tile = blockIdx.x * 4 + wave;          // 0..1023
  int tj = tile >> 4;                        // 64 row tiles (P/16)
  int td = tile & 15;                        // 16 col tiles (D/16)
  int j0 = tj * 16;
  int d0 = td * 16;
  int m = lane & 15;
  int half = lane >> 4;

  v8f acc = {};
  const float* arow = E + (size_t)(j0 + m) * PP;  // A row: E[j0+m, :]
  const float* bcol = Pr + (d0 + m);              // B col: Proto[:, d0+n]

  for (int k = 0; k < PP; k += 4) {
    int kb = k + 2 * half;
    v2f a = *(const v2f*)(arow + kb);             // A[m, kb], A[m, kb+1]
    v2f b;
    b.x = bcol[(size_t)kb * DD];                  // B[kb,   n]
    b.y = bcol[(size_t)(kb + 1) * DD];            // B[kb+1, n]
    acc = __builtin_amdgcn_wmma_f32_16x16x4_f32(false, a, false, b,
                                                (short)0, acc, false, false);
  }

  // D layout: vgpr v -> row j0 + v + 8*half, col d0 + (lane&15).
  int d = d0 + m;
#pragma unroll
  for (int v = 0; v < 8; ++v) {
    int j = j0 + v + 8 * half;
    Mt[(size_t)d * PP + j] = acc[v] * inv_r[j];
  }
}

// ---------------------------------------------------------------------------
// Kernel 5 (WMMA + fused min): out[n] = 0.5*(x_sq[n] + min_j(b[j] - 2*S[n,j]))
// with S = X @ Mt  (A = X: N x D from LDS, B = Mt: D x P).
// One block (8 waves) per 16 samples; wave w handles column tiles w, w+8, ...
// ---------------------------------------------------------------------------
__global__ __launch_bounds__(256) void som_gemm2_min(const float* __restrict__ X,
                                                     const float* __restrict__ Mt,
                                                     const float* __restrict__ bv,
                                                     const float* __restrict__ x_sq,
                                                     float* __restrict__ out) {
  __shared__ float lds_x[16 * DD];        // 16 KB: 16 sample rows
  __shared__ float lds_min[8][16];

  int n0 = blockIdx.x * 16;
  for (int i = threadIdx.x; i < 16 * DD; i += 256)
    lds_x[i] = X[(size_t)(n0 + (i >> 8)) * DD + (i & (DD - 1))];
  __syncthreads();

  int wave = threadIdx.x >> 5;
  int lane = threadIdx.x & 31;
  int m = lane & 15;
  int half = lane >> 4;

  float best[8];
#pragma unroll
  for (int v = 0; v < 8; ++v) best[v] = 3.4e38f;

  for (int tjt = wave; tjt < PP / 16; tjt += 8) {
    int j0 = tjt * 16;
    v8f acc = {};
    const float* arow = lds_x + m * DD;     // A row: X[n0+m, :] (LDS)
    const float* bcol = Mt + (j0 + m);      // B col: Mt[:, j0+n]
    for (int k = 0; k < DD; k += 4) {
      int kb = k + 2 * half;
      v2f a = *(const v2f*)(arow + kb);
      v2f b;
      b.x = bcol[(size_t)kb * PP];
      b.y = bcol[(size_t)(kb + 1) * PP];
      acc = __builtin_amdgcn_wmma_f32_16x16x4_f32(false, a, false, b,
                                                  (short)0, acc, false, false);
    }
    float bj = bv[j0 + m];                  // this lane's column j
#pragma unroll
    for (int v = 0; v < 8; ++v) {
      float t = bj - 2.0f * acc[v];
      best[v] = fminf(best[v], t);
    }
  }

  // min across the 16 lanes of each half (same output row, different j)
#pragma unroll
  for (int off = 1; off < 16; off <<= 1) {
#pragma unroll
    for (int v = 0; v < 8; ++v)
      best[v] = fminf(best[v], __shfl_xor(best[v], off, 32));
  }
  if (m == 0) {  // lanes 0 and 16: rows half*8 + v
#pragma unroll
    for (int v = 0; v < 8; ++v) lds_min[wave][half * 8 + v] = best[v];
  }
  __syncthreads();

  if (threadIdx.x < 16) {
    float mn = lds_min[0][threadIdx.x];
#pragma unroll
    for (int w = 1; w < 8; ++w) mn = fminf(mn, lds_min[w][threadIdx.x]);
    int n = n0 + threadIdx.x;
    out[n] = 0.5f * (x_sq[n] + mn);
  }
}

// ---------------------------------------------------------------------------
extern "C" void kernel_launch(void* const* d_in, const int* in_sizes, int n_in,
                              void* d_out, int out_size, void* d_ws, size_t ws_size,
                              hipStream_t stream) {
  (void)in_sizes; (void)n_in; (void)out_size; (void)ws_size;

  const float* X     = (const float*)d_in[0];   // (N, D)
  const float* Pr    = (const float*)d_in[1];   // (P, D)
  const float* dist  = (const float*)d_in[2];   // (P, P)
  const float* sigma = (const float*)d_in[3];   // (1,)
  float* out = (float*)d_out;                   // (N,)

  // Workspace layout (floats): E | Mt | x_sq | p_sq | inv_r | b  (~5.3 MB)
  float* ws    = (float*)d_ws;
  float* E     = ws;                              // P*P
  float* Mt    = E + (size_t)PP * PP;             // D*P (M transposed)
  float* x_sq  = Mt + (size_t)DD * PP;            // N
  float* p_sq  = x_sq + NS;                       // P
  float* inv_r = p_sq + PP;                       // P
  float* bvec  = inv_r + PP;                      // P

  som_sq_norms<<<NS + PP, 256, 0, stream>>>(X, Pr, x_sq, p_sq);
  som_exp<<<(PP * PP) / 256, 256, 0, stream>>>(dist, sigma, E);
  som_row_stats<<<PP, 256, 0, stream>>>(E, p_sq, inv_r, bvec);
  som_gemm1<<<(PP / 16) * (DD / 16) / 4, 128, 0, stream>>>(E, Pr, inv_r, Mt);
  som_gemm2_min<<<NS / 16, 256, 0, stream>>>(X, Mt, bvec, x_sq, out);
}